// FNO1D_71614284693696
// MI455X (gfx1250) — compile-verified
//
#include <hip/hip_runtime.h>
#include <hip/hip_bf16.h>

// ---------------------------------------------------------------------------
// FNO1D forward for MI455X (gfx1250), wave32 + WMMA f32 16x16x4.
// Heavy ops are GEMMs over V_WMMA_F32_16X16X4_F32; LDS staging uses the
// CDNA5 async global->LDS path (ASYNCcnt) when the builtin is available.
// ---------------------------------------------------------------------------

#define B_   64
#define L_   4096
#define W_   128
#define M_   64     // retained Fourier modes
#define NF_  128    // 64 real + 64 imag coefficients
#define E_   128    // time-embedding dim

typedef __attribute__((ext_vector_type(2))) float v2f;
typedef __attribute__((ext_vector_type(8))) float v8f;

__device__ __forceinline__ v8f wmma_acc(v2f a, v2f b, v8f c) {
  // ISA: for F32 shapes NEG = {CNeg,0,0}; A/B neg modifiers must be 0.
  return __builtin_amdgcn_wmma_f32_16x16x4_f32(false, a, false, b, (short)0, c,
                                               false, false);
}

// ---- async global->LDS 16-byte copy (GLOBAL_LOAD_ASYNC_TO_LDS_B128) -------
// Builtin signature (from hipcc diagnostic): param 1 is
// 'int __vector(4) __attribute__((address_space(1)))*' (global src),
// param 2 LDS dst, then two immediate ints (offset, cpol).
typedef int v4i_ __attribute__((vector_size(16)));
typedef __attribute__((address_space(1))) v4i_* gv4_ptr;
typedef __attribute__((address_space(3))) v4i_* lv4_ptr;

#if __has_builtin(__builtin_amdgcn_global_load_async_to_lds_b128) && \
    __has_builtin(__builtin_amdgcn_s_wait_asynccnt)
__device__ __forceinline__ void async_copy16(void* lds, const void* g) {
  __builtin_amdgcn_global_load_async_to_lds_b128((gv4_ptr)g, (lv4_ptr)lds, 0, 0);
}
__device__ __forceinline__ void async_wait0() {
  __builtin_amdgcn_s_wait_asynccnt(0);
}
#else
__device__ __forceinline__ void async_copy16(void* lds, const void* g) {
  *(float4*)lds = *(const float4*)g;
}
__device__ __forceinline__ void async_wait0() {}
#endif

__device__ __forceinline__ float gelu_exact(float v) {
  return 0.5f * v * (1.0f + erff(v * 0.7071067811865476f));
}
__device__ __forceinline__ float silu(float v) {
  return v / (1.0f + __expf(-v));
}

// ---------------------------------------------------------------------------
// Basis tables: fwd DFT [L][128] (cos | -sin), inverse [128][L] (scaled).
// ---------------------------------------------------------------------------
__global__ void k_basis(float* __restrict__ fwdB, float* __restrict__ invB) {
  int idx = blockIdx.x * 256 + threadIdx.x;  // over L_*M_
  if (idx >= L_ * M_) return;
  int l = idx >> 6;
  int k = idx & 63;
  int r = (k * l) & (L_ - 1);  // phase reduced exactly mod L (L = 2^12)
  float ang = (6.283185307179586f / (float)L_) * (float)r;
  float s = __sinf(ang), c = __cosf(ang);
  // rfft: X_k = sum h[l] * e^{-i*ang}:  Re += h*cos, Im += -h*sin
  fwdB[l * NF_ + k] = c;
  fwdB[l * NF_ + 64 + k] = -s;
  // irfft of 64-mode spectrum:
  // x[l] = (1/L)*ReX0 + (2/L)*sum_{k>=1}(ReXk*cos - ImXk*sin)
  float sc = (k == 0) ? (1.0f / (float)L_) : (2.0f / (float)L_);
  invB[k * L_ + l] = sc * c;
  invB[(64 + k) * L_ + l] = (k == 0) ? 0.0f : -sc * s;
}

// ---------------------------------------------------------------------------
// Sinusoidal time embedding -> Linear -> SiLU.  grid=B, block=128.
// ---------------------------------------------------------------------------
__global__ void k_temb(const float* __restrict__ t, const float* __restrict__ tW,
                       const float* __restrict__ tb, float* __restrict__ temb) {
  __shared__ float emb[E_];
  int b = blockIdx.x, o = threadIdx.x;
  float tv = t[b];
  int j = (o < 64) ? o : (o - 64);
  float f = __expf(-(float)j * (9.210340371976184f / 63.0f));  // ln(1e4)/63
  emb[o] = (o < 64) ? __sinf(tv * f) : __cosf(tv * f);
  __syncthreads();
  float acc = tb[o];
  const float* wrow = tW + o * E_;
#pragma unroll 8
  for (int jj = 0; jj < E_; ++jj) acc += emb[jj] * wrow[jj];
  temb[b * E_ + o] = silu(acc);
}

// ---------------------------------------------------------------------------
// Per-layer time MLP: te = silu(temb@W1^T+b1)@W2^T+b2.  grid=B, block=128.
// ---------------------------------------------------------------------------
__global__ void k_te(const float* __restrict__ temb, const float* __restrict__ W1,
                     const float* __restrict__ b1, const float* __restrict__ W2,
                     const float* __restrict__ b2, float* __restrict__ te) {
  __shared__ float u[W_];
  int b = blockIdx.x, o = threadIdx.x;
  const float* tbv = temb + b * E_;
  float acc = b1[o];
  const float* w1r = W1 + o * E_;
#pragma unroll 8
  for (int j = 0; j < E_; ++j) acc += tbv[j] * w1r[j];
  u[o] = silu(acc);
  __syncthreads();
  float acc2 = b2[o];
  const float* w2r = W2 + o * W_;
#pragma unroll 8
  for (int j = 0; j < W_; ++j) acc2 += u[j] * w2r[j];
  te[b * W_ + o] = acc2;
}

// ---------------------------------------------------------------------------
// Lift: h[b,w,l] = fc0_W[w]*x[b,l] + fc0_b[w].  One element per thread.
// ---------------------------------------------------------------------------
__global__ void k_lift(const float* __restrict__ x, const float* __restrict__ w,
                       const float* __restrict__ bias, float* __restrict__ h) {
  long idx = (long)blockIdx.x * 256 + threadIdx.x;  // < B*W*L = 2^25
  int l = (int)(idx & (L_ - 1));
  int wc = (int)((idx >> 12) & 127);
  int b = (int)(idx >> 19);
  h[idx] = w[wc] * x[(long)b * L_ + l] + bias[wc];
}

// ---------------------------------------------------------------------------
// Forward DFT (64 modes): C[8192x128] = h[8192x4096] * fwdB[4096x128].
// Output stored column-major: hft[col*8192 + row].
// grid = 512 row-stripes, block = 256 (8 waves, wave -> one 16-col tile).
// K-panels staged to LDS via async b128 copies.
// ---------------------------------------------------------------------------
__global__ __launch_bounds__(256) void k_fwddft(const float* __restrict__ h,
                                                const float* __restrict__ fwdB,
                                                float* __restrict__ hft) {
  __shared__ __align__(16) float As[16 * 64];
  __shared__ __align__(16) float Fs[64 * 128];
  const int tid = threadIdx.x;
  const int lane = tid & 31, wave = tid >> 5;
  const int half = lane >> 4, m = lane & 15;
  const int row0 = blockIdx.x * 16;
  const int ncol = wave * 16 + m;
  v8f acc = {};
  for (int kk = 0; kk < L_; kk += 64) {
    __syncthreads();  // previous panel fully consumed
    {  // A panel: 16x64 = 1024 floats = 256 x b128 (1 per thread)
      int i = tid * 4;
      int r = i >> 6, c = i & 63;
      async_copy16(&As[i], h + (long)(row0 + r) * L_ + kk + c);
    }
#pragma unroll
    for (int q = 0; q < 8; ++q) {  // F panel: 64x128 = 2048 x b128
      int i = (q * 256 + tid) * 4;
      int r = i >> 7, c = i & 127;
      async_copy16(&Fs[i], fwdB + (long)(kk + r) * NF_ + c);
    }
    async_wait0();
    __syncthreads();
#pragma unroll
    for (int k4 = 0; k4 < 64; k4 += 4) {
      v2f a, bv;
      a.x = As[m * 64 + k4 + 2 * half];
      a.y = As[m * 64 + k4 + 2 * half + 1];
      bv.x = Fs[(k4 + 2 * half) * 128 + ncol];
      bv.y = Fs[(k4 + 2 * half + 1) * 128 + ncol];
      acc = wmma_acc(a, bv, acc);
    }
  }
  float* dst = hft + (long)ncol * 8192 + row0 + 8 * half;
#pragma unroll
  for (int r = 0; r < 8; ++r) dst[r] = acc[r];
}

// ---------------------------------------------------------------------------
// Spectral multiply, per mode k: (Hr+iHi)[64x128] x (Wr+iWi)[128x128].
// grid = 64 modes, block = 256 (8 waves x 4 tiles each = 32 tiles of 16x16).
// The -Hi*Wi term uses a VALU-negated A operand (f32 WMMA has no A-neg mod).
// ---------------------------------------------------------------------------
__global__ __launch_bounds__(256) void k_specmul(const float* __restrict__ hft,
                                                 const float* __restrict__ wr,
                                                 const float* __restrict__ wi,
                                                 float* __restrict__ ol) {
  const int k = blockIdx.x;
  const int tid = threadIdx.x, lane = tid & 31, wave = tid >> 5;
  const int half = lane >> 4, m = lane & 15;
  const float* Hr = hft + (long)k * 8192;
  const float* Hi = hft + (long)(64 + k) * 8192;
  for (int t = wave * 4; t < wave * 4 + 4; ++t) {
    const int mt = t >> 3, nt = t & 7;
    const int o = nt * 16 + m;       // B-matrix column (out channel)
    const int brow = mt * 16 + m;    // A-matrix row (batch)
    v8f accR = {}, accI = {};
#pragma unroll 4
    for (int k4 = 0; k4 < 128; k4 += 4) {
      int ka = k4 + 2 * half;
      v2f ar, ai, nai, br, bi;
      ar.x = Hr[brow * 128 + ka];     ar.y = Hr[brow * 128 + ka + 1];
      ai.x = Hi[brow * 128 + ka];     ai.y = Hi[brow * 128 + ka + 1];
      nai.x = -ai.x;                  nai.y = -ai.y;
      br.x = wr[(long)ka * 8192 + o * 64 + k];
      br.y = wr[(long)(ka + 1) * 8192 + o * 64 + k];
      bi.x = wi[(long)ka * 8192 + o * 64 + k];
      bi.y = wi[(long)(ka + 1) * 8192 + o * 64 + k];
      accR = wmma_acc(ar, br, accR);
      accR = wmma_acc(nai, bi, accR);  // Re -= Hi*Wi
      accI = wmma_acc(ar, bi, accI);
      accI = wmma_acc(ai, br, accI);
    }
#pragma unroll
    for (int r = 0; r < 8; ++r) {
      int row = mt * 16 + r + 8 * half;  // batch index
      ol[(long)k * 8192 + row * 128 + o] = accR[r];
      ol[(long)(64 + k) * 8192 + row * 128 + o] = accI[r];
    }
  }
}

// ---------------------------------------------------------------------------
// Inverse DFT: x1[8192x4096] = ol[8192x128] * invB[128x4096].
// ol is column-major ([coeff][row]); A panel staged to LDS as [coeff][row]
// so both async copies are contiguous 16B runs.
// grid = (512 row-stripes, 32 l-chunks of 128), block = 256.
// ---------------------------------------------------------------------------
__global__ __launch_bounds__(256) void k_idft(const float* __restrict__ ol,
                                              const float* __restrict__ invB,
                                              float* __restrict__ x1) {
  __shared__ __align__(16) float As[128 * 16];  // [coeff j][row r]
  __shared__ __align__(16) float Fs[64 * 128];
  const int tid = threadIdx.x, lane = tid & 31, wave = tid >> 5;
  const int half = lane >> 4, m = lane & 15;
  const int row0 = blockIdx.x * 16;
  const int nbase = blockIdx.y * 128;
#pragma unroll
  for (int q = 0; q < 2; ++q) {  // A panel: 128x16 = 512 x b128
    int i = (q * 256 + tid) * 4;
    int j = i >> 4, r = i & 15;
    async_copy16(&As[i], ol + (long)j * 8192 + row0 + r);
  }
  v8f acc = {};
  for (int kk = 0; kk < 128; kk += 64) {
    __syncthreads();
#pragma unroll
    for (int q = 0; q < 8; ++q) {  // F panel: 64x128 = 2048 x b128
      int i = (q * 256 + tid) * 4;
      int r = i >> 7, c = i & 127;
      async_copy16(&Fs[i], invB + (long)(kk + r) * L_ + nbase + c);
    }
    async_wait0();
    __syncthreads();
#pragma unroll
    for (int k4 = 0; k4 < 64; k4 += 4) {
      int ka = k4 + 2 * half;
      v2f a, bv;
      a.x = As[(kk + ka) * 16 + m];
      a.y = As[(kk + ka + 1) * 16 + m];
      bv.x = Fs[ka * 128 + wave * 16 + m];
      bv.y = Fs[(ka + 1) * 128 + wave * 16 + m];
      acc = wmma_acc(a, bv, acc);
    }
  }
  const int col = nbase + wave * 16 + m;
#pragma unroll
  for (int r = 0; r < 8; ++r) {
    int row = row0 + r + 8 * half;
    x1[(long)row * L_ + col] = acc[r];
  }
}

// ---------------------------------------------------------------------------
// 1x1 conv + residual fuse: h' = gelu(W*h + b + te + x1).
// x1 lives in hx; result overwritten into hx (same lane reads then writes).
// grid = (B, 256 l-tiles), block = 256 (8 waves cover 128 out channels).
// ---------------------------------------------------------------------------
__global__ __launch_bounds__(256) void k_conv(const float* __restrict__ h,
                                              float* __restrict__ hx,
                                              const float* __restrict__ wW,
                                              const float* __restrict__ wb,
                                              const float* __restrict__ te) {
  __shared__ __align__(16) float Bs[128 * 16];
  const int tid = threadIdx.x, lane = tid & 31, wave = tid >> 5;
  const int half = lane >> 4, m = lane & 15;
  const int b = blockIdx.x;
  const int l0 = blockIdx.y * 16;
  const float* hb = h + (long)b * W_ * L_;
#pragma unroll
  for (int q = 0; q < 2; ++q) {  // B panel: 128x16 = 512 x b128
    int i = (q * 256 + tid) * 4;
    int ch = i >> 4, n = i & 15;
    async_copy16(&Bs[i], hb + (long)ch * L_ + l0 + n);
  }
  async_wait0();
  __syncthreads();
  const int o0 = wave * 16;
  v8f acc = {};
#pragma unroll 8
  for (int k4 = 0; k4 < 128; k4 += 4) {
    int ka = k4 + 2 * half;
    v2f a, bv;
    a.x = wW[(o0 + m) * 128 + ka];
    a.y = wW[(o0 + m) * 128 + ka + 1];
    bv.x = Bs[ka * 16 + m];
    bv.y = Bs[(ka + 1) * 16 + m];
    acc = wmma_acc(a, bv, acc);
  }
  const int col = l0 + m;
  float* hxb = hx + (long)b * W_ * L_;
#pragma unroll
  for (int r = 0; r < 8; ++r) {
    int o = o0 + r + 8 * half;
    float v = acc[r] + wb[o] + te[b * W_ + o] + hxb[(long)o * L_ + col];
    hxb[(long)o * L_ + col] = gelu_exact(v);
  }
}

// ---------------------------------------------------------------------------
// Projection: y[b,l] = fc2_b + sum_o fc2[o]*gelu(fc1[o,:].h[b,:,l] + fc1_b[o]).
// grid = (B, 256 l-tiles), block = 256; LDS reduction over the 128 channels.
// ---------------------------------------------------------------------------
__global__ __launch_bounds__(256) void k_proj(const float* __restrict__ h,
                                              const float* __restrict__ f1W,
                                              const float* __restrict__ f1b,
                                              const float* __restrict__ f2W,
                                              const float* __restrict__ f2b,
                                              float* __restrict__ out) {
  __shared__ __align__(16) float Bs[128 * 16];
  __shared__ float red[16];
  const int tid = threadIdx.x, lane = tid & 31, wave = tid >> 5;
  const int half = lane >> 4, m = lane & 15;
  const int b = blockIdx.x;
  const int l0 = blockIdx.y * 16;
  const float* hb = h + (long)b * W_ * L_;
#pragma unroll
  for (int q = 0; q < 2; ++q) {
    int i = (q * 256 + tid) * 4;
    int ch = i >> 4, n = i & 15;
    async_copy16(&Bs[i], hb + (long)ch * L_ + l0 + n);
  }
  if (tid < 16) red[tid] = 0.0f;
  async_wait0();
  __syncthreads();
  const int o0 = wave * 16;
  v8f acc = {};
#pragma unroll 8
  for (int k4 = 0; k4 < 128; k4 += 4) {
    int ka = k4 + 2 * half;
    v2f a, bv;
    a.x = f1W[(o0 + m) * 128 + ka];
    a.y = f1W[(o0 + m) * 128 + ka + 1];
    bv.x = Bs[ka * 16 + m];
    bv.y = Bs[(ka + 1) * 16 + m];
    acc = wmma_acc(a, bv, acc);
  }
  float p = 0.0f;
#pragma unroll
  for (int r = 0; r < 8; ++r) {
    int o = o0 + r + 8 * half;
    p += f2W[o] * gelu_exact(acc[r] + f1b[o]);
  }
  atomicAdd(&red[m], p);
  __syncthreads();
  if (tid < 16) out[(long)b * L_ + l0 + tid] = red[tid] + f2b[0];
}

// ---------------------------------------------------------------------------
// Host launcher.
// Input order: x, t, time_W, time_b, fc0_W, fc0_b,
//   [spec_wr, spec_wi, w_W, w_b, tm_W1, tm_b1, tm_W2, tm_b2] x 4,
//   fc1_W, fc1_b, fc2_W, fc2_b.
// Workspace (floats): temb 8K | te 8K | fwdB 512K | invB 512K |
//   hft 1M | olow 1M | hA 32M | hB 32M   (~281 MB total)
// ---------------------------------------------------------------------------
extern "C" void kernel_launch(void* const* d_in, const int* in_sizes, int n_in,
                              void* d_out, int out_size, void* d_ws,
                              size_t ws_size, hipStream_t stream) {
  (void)in_sizes; (void)n_in; (void)out_size; (void)ws_size;
  const float* x      = (const float*)d_in[0];
  const float* t      = (const float*)d_in[1];
  const float* time_W = (const float*)d_in[2];
  const float* time_b = (const float*)d_in[3];
  const float* fc0_W  = (const float*)d_in[4];
  const float* fc0_b  = (const float*)d_in[5];
  const float* fc1_W  = (const float*)d_in[38];
  const float* fc1_b  = (const float*)d_in[39];
  const float* fc2_W  = (const float*)d_in[40];
  const float* fc2_b  = (const float*)d_in[41];

  float* ws   = (float*)d_ws;
  float* temb = ws;                         // 8192
  float* te   = temb + 8192;                // 8192
  float* fwdB = te + 8192;                  // 4096*128
  float* invB = fwdB + (size_t)L_ * NF_;    // 128*4096
  float* hft  = invB + (size_t)NF_ * L_;    // 128*8192
  float* olow = hft + (size_t)NF_ * 8192;   // 128*8192
  float* hA   = olow + (size_t)NF_ * 8192;  // 64*128*4096
  float* hB   = hA + (size_t)B_ * W_ * L_;  // 64*128*4096

  k_basis<<<(L_ * M_ + 255) / 256, 256, 0, stream>>>(fwdB, invB);
  k_temb<<<B_, 128, 0, stream>>>(t, time_W, time_b, temb);
  k_lift<<<(B_ * W_ * L_) / 256, 256, 0, stream>>>(x, fc0_W, fc0_b, hA);

  float* hcur = hA;
  float* hoth = hB;
  for (int i = 0; i < 4; ++i) {
    const float* wr   = (const float*)d_in[6 + 8 * i + 0];
    const float* wi   = (const float*)d_in[6 + 8 * i + 1];
    const float* wW   = (const float*)d_in[6 + 8 * i + 2];
    const float* wb   = (const float*)d_in[6 + 8 * i + 3];
    const float* tmW1 = (const float*)d_in[6 + 8 * i + 4];
    const float* tmb1 = (const float*)d_in[6 + 8 * i + 5];
    const float* tmW2 = (const float*)d_in[6 + 8 * i + 6];
    const float* tmb2 = (const float*)d_in[6 + 8 * i + 7];

    k_te<<<B_, 128, 0, stream>>>(temb, tmW1, tmb1, tmW2, tmb2, te);
    k_fwddft<<<512, 256, 0, stream>>>(hcur, fwdB, hft);
    k_specmul<<<M_, 256, 0, stream>>>(hft, wr, wi, olow);
    k_idft<<<dim3(512, 32), 256, 0, stream>>>(olow, invB, hoth);
    k_conv<<<dim3(B_, L_ / 16), 256, 0, stream>>>(hcur, hoth, wW, wb, te);
    float* tmp = hcur; hcur = hoth; hoth = tmp;
  }

  k_proj<<<dim3(B_, L_ / 16), 256, 0, stream>>>(hcur, fc1_W, fc1_b, fc2_W,
                                                fc2_b, (float*)d_out);
}